// ProcrustesAlignment_21285857919534
// MI455X (gfx1250) — compile-verified
//
#include <hip/hip_runtime.h>

typedef __attribute__((ext_vector_type(2))) float v2f;
typedef __attribute__((ext_vector_type(4))) float v4f;
typedef __attribute__((ext_vector_type(8))) float v8f;

#define BM 64
#define BN 64
#define BK 16
#define LDSS 20   // LDS row stride (floats): 80B rows -> b128-aligned stores, b64-aligned frag reads

// ---------------------------------------------------------------------------
// Tiled batched GEMM on the fp32 WMMA pipe (no bounds checks; requires
// M%64==0, N%64==0, K%16==0 — guaranteed after rank padding 24->32):
//   D[b] = alpha * op(A[b]) @ op(B[b]) + beta * Cin[b] + diag * I
// 128 threads = 4 waves; C tile 64x64; each wave computes a 16x64 strip
// (4 accumulators). A: LDS row-major [m][k]; B: LDS K-transposed [n][k] so
// both fragment pairs are contiguous ds_load_b64.
// ---------------------------------------------------------------------------
template<bool TA, bool TB>
__global__ __launch_bounds__(128)
void gemm_wmma_tiled(const float* __restrict__ A, const float* __restrict__ Bp,
                     const float* __restrict__ Cin, float* __restrict__ D,
                     int M, int N, int K, int lda, int ldb, int ldci, int ldco,
                     long long sA, long long sB, long long sCi, long long sCo,
                     float alpha, float beta, float diag)
{
    __shared__ float As[BM * LDSS];
    __shared__ float Bs[BN * LDSS];

    const int bt = blockIdx.z;
    const float* Ab = A   + (long long)bt * sA;
    const float* Bb = Bp  + (long long)bt * sB;
    const float* Cb = Cin + (long long)bt * sCi;
    float*       Db = D   + (long long)bt * sCo;

    const int nt = blockIdx.x * BN;
    const int mt = blockIdx.y * BM;

    const int tid  = threadIdx.x;      // 0..127
    const int wave = tid >> 5;         // 0..3
    const int lane = tid & 31;
    const int half = lane >> 4;
    const int r    = lane & 15;

    v8f acc[4] = {};

    for (int k0 = 0; k0 < K; k0 += BK) {
        __syncthreads();
        // ---- stage A tile: As[m][k] = opA(mt+m, k0+k)
        if (TA) {       // physical A is KxM: 16 rows x 64 cols, coalesced on m
            const int k  = tid >> 3;
            const int mg = (tid & 7) * 8;
            const float* g = Ab + (long long)(k0 + k) * lda + (mt + mg);
            v4f x0 = *(const v4f*)g;
            v4f x1 = *(const v4f*)(g + 4);
            if (k0 + BK < K) __builtin_prefetch(g + (long long)BK * lda, 0, 1);
#pragma unroll
            for (int j = 0; j < 4; ++j) As[(mg + j) * LDSS + k] = x0[j];
#pragma unroll
            for (int j = 0; j < 4; ++j) As[(mg + 4 + j) * LDSS + k] = x1[j];
        } else {        // physical A is MxK: 64 rows x 16 cols
            const int m  = tid >> 1;
            const int kg = (tid & 1) * 8;
            const float* g = Ab + (long long)(mt + m) * lda + (k0 + kg);
            if (k0 + BK < K) __builtin_prefetch(g + BK, 0, 1);
            *(v4f*)&As[m * LDSS + kg]     = *(const v4f*)g;
            *(v4f*)&As[m * LDSS + kg + 4] = *(const v4f*)(g + 4);
        }
        // ---- stage B tile K-transposed: Bs[n][k] = opB(k0+k, nt+n)
        if (TB) {       // physical B is NxK: 64 rows x 16 cols -> direct
            const int nb = tid >> 1;
            const int kg = (tid & 1) * 8;
            const float* g = Bb + (long long)(nt + nb) * ldb + (k0 + kg);
            if (k0 + BK < K) __builtin_prefetch(g + BK, 0, 1);
            *(v4f*)&Bs[nb * LDSS + kg]     = *(const v4f*)g;
            *(v4f*)&Bs[nb * LDSS + kg + 4] = *(const v4f*)(g + 4);
        } else {        // physical B is KxN: 16 rows x 64 cols -> transpose
            const int k  = tid >> 3;
            const int ng = (tid & 7) * 8;
            const float* g = Bb + (long long)(k0 + k) * ldb + (nt + ng);
            v4f x0 = *(const v4f*)g;
            v4f x1 = *(const v4f*)(g + 4);
            if (k0 + BK < K) __builtin_prefetch(g + (long long)BK * ldb, 0, 1);
#pragma unroll
            for (int j = 0; j < 4; ++j) Bs[(ng + j) * LDSS + k] = x0[j];
#pragma unroll
            for (int j = 0; j < 4; ++j) Bs[(ng + 4 + j) * LDSS + k] = x1[j];
        }
        __syncthreads();
        // ---- compute: 16 WMMAs per K-step per wave
#pragma unroll
        for (int kk = 0; kk < BK; kk += 4) {
            const int ka = kk + 2 * half;
            v2f av = *(const v2f*)&As[(wave * 16 + r) * LDSS + ka];
#pragma unroll
            for (int t = 0; t < 4; ++t) {
                v2f bv = *(const v2f*)&Bs[(t * 16 + r) * LDSS + ka];
                acc[t] = __builtin_amdgcn_wmma_f32_16x16x4_f32(
                    false, av, false, bv, (short)0, acc[t], false, false);
            }
        }
    }
    // ---- epilogue
#pragma unroll
    for (int t = 0; t < 4; ++t) {
        const int nn = nt + t * 16 + r;
#pragma unroll
        for (int v = 0; v < 8; ++v) {
            const int mm = mt + wave * 16 + v + 8 * half;
            float val = alpha * acc[t][v];
            if (beta != 0.f) val += beta * Cb[(long long)mm * ldci + nn];
            if (mm == nn)    val += diag;
            Db[(long long)mm * ldco + nn] = val;
        }
    }
}

// ---------------------------------------------------------------------------
// Simple branch-free GEMM for the small 32-wide panels (dims % 16 == 0).
// One wave per 16x16 tile.
// ---------------------------------------------------------------------------
template<bool TA, bool TB>
__global__ __launch_bounds__(32)
void gemm_wmma_simple(const float* __restrict__ A, const float* __restrict__ Bp,
                      const float* __restrict__ Cin, float* __restrict__ D,
                      int M, int N, int K, int lda, int ldb, int ldci, int ldco,
                      long long sA, long long sB, long long sCi, long long sCo,
                      float alpha, float beta, float diag)
{
    const int bt = blockIdx.z;
    const float* Ab = A   + (long long)bt * sA;
    const float* Bb = Bp  + (long long)bt * sB;
    const float* Cb = Cin + (long long)bt * sCi;
    float*       Db = D   + (long long)bt * sCo;

    const int nt   = blockIdx.x * 16;
    const int mt   = blockIdx.y * 16;
    const int lane = threadIdx.x & 31;
    const int half = lane >> 4;
    const int r    = lane & 15;

    const int m  = mt + r;
    const int nn = nt + r;
    const int ko = 2 * half;

    v8f acc = {};
#pragma unroll 4
    for (int k0 = 0; k0 < K; k0 += 4) {
        const int ka = k0 + ko;
        v2f av, bv;
        if (TA) {
            av.x = Ab[(long long)ka * lda + m];
            av.y = Ab[(long long)(ka + 1) * lda + m];
        } else {
            av = *(const v2f*)&Ab[(long long)m * lda + ka];
        }
        if (TB) {
            bv = *(const v2f*)&Bb[(long long)nn * ldb + ka];
        } else {
            bv.x = Bb[(long long)ka * ldb + nn];
            bv.y = Bb[(long long)(ka + 1) * ldb + nn];
        }
        acc = __builtin_amdgcn_wmma_f32_16x16x4_f32(
            false, av, false, bv, (short)0, acc, false, false);
    }
#pragma unroll
    for (int v = 0; v < 8; ++v) {
        const int mm = mt + v + 8 * half;
        float val = alpha * acc[v];
        if (beta != 0.f) val += beta * Cb[(long long)mm * ldci + nn];
        if (mm == nn)    val += diag;
        Db[(long long)mm * ldco + nn] = val;
    }
}

// ---------------------------------------------------------------------------
// Small helper kernels
// ---------------------------------------------------------------------------
__global__ __launch_bounds__(256)
void frob_kernel(const float* __restrict__ X, long long stride, int count,
                 float* __restrict__ out)
{
    const int b = blockIdx.x;
    const float* Xb = X + (long long)b * stride;
    float s = 0.f;
    for (int i = threadIdx.x; i < count; i += 256) { float v = Xb[i]; s += v * v; }
    __shared__ float red[256];
    red[threadIdx.x] = s; __syncthreads();
    for (int off = 128; off > 0; off >>= 1) {
        if (threadIdx.x < off) red[threadIdx.x] += red[threadIdx.x + off];
        __syncthreads();
    }
    if (threadIdx.x == 0) out[b] = sqrtf(red[0]);
}

// Y = X * f(s[b]); mode 0: f = 1/s, mode 1: f = 1/sqrt(s)
__global__ __launch_bounds__(256)
void scale_kernel(const float* __restrict__ X, float* __restrict__ Y,
                  const float* __restrict__ s, int mode, long long total, int per)
{
    long long i = (long long)blockIdx.x * 256 + threadIdx.x;
    if (i >= total) return;
    const int b = (int)(i / per);
    float sc = s[b];
    sc = (mode == 0) ? (1.f / sc) : rsqrtf(sc);
    Y[i] = X[i] * sc;
}

__global__ __launch_bounds__(256)
void set_diag_kernel(float* __restrict__ Y, float dval, int dim,
                     long long total, int per)
{
    long long i = (long long)blockIdx.x * 256 + threadIdx.x;
    if (i >= total) return;
    const int idx = (int)(i % per);
    Y[i] = ((idx / dim) == (idx % dim)) ? dval : 0.f;
}

__global__ __launch_bounds__(256)
void shift_diag_kernel(const float* __restrict__ X, float* __restrict__ Y,
                       float dval, int dim, long long total, int per)
{
    long long i = (long long)blockIdx.x * 256 + threadIdx.x;
    if (i >= total) return;
    const int idx = (int)(i % per);
    Y[i] = X[i] + (((idx / dim) == (idx % dim)) ? dval : 0.f);
}

// Deterministic seed for subspace iteration; columns >= rank stay exactly 0
// so the 24->32 padding is preserved through the whole pipeline.
__global__ __launch_bounds__(256)
void qinit_kernel(float* __restrict__ Q, long long total, int ld, int rank)
{
    long long i = (long long)blockIdx.x * 256 + threadIdx.x;
    if (i >= total) return;
    const int col = (int)(i % ld);
    float val = 0.f;
    if (col < rank) {
        unsigned h = (unsigned)i * 2654435761u ^ 0x9E3779B9u;
        h ^= h >> 16; h *= 0x85EBCA6Bu; h ^= h >> 13; h *= 0xC2B2AE35u; h ^= h >> 16;
        val = ((float)(h & 0xFFFFu) * (1.0f / 65536.0f)) - 0.5f;
    }
    Q[i] = val;
}

__global__ __launch_bounds__(256)
void cos_partial_kernel(const float* __restrict__ aligned,
                        const float* __restrict__ target,
                        float* __restrict__ partial, long long rows, int N)
{
    long long row = (long long)blockIdx.x * 256 + threadIdx.x;
    float c = 0.f;
    if (row < rows) {
        const float* a = aligned + row * N;
        const float* t = target  + row * N;
        float dot = 0.f, na = 0.f, ntn = 0.f;
        for (int i = 0; i < N; ++i) {
            float av = a[i], tv = t[i];
            dot += av * tv; na += av * av; ntn += tv * tv;
        }
        c = dot / (sqrtf(na) * sqrtf(ntn) + 1e-8f);
    }
    __shared__ float red[256];
    red[threadIdx.x] = c; __syncthreads();
    for (int off = 128; off > 0; off >>= 1) {
        if (threadIdx.x < off) red[threadIdx.x] += red[threadIdx.x + off];
        __syncthreads();
    }
    if (threadIdx.x == 0) partial[blockIdx.x] = red[0];
}

__global__ __launch_bounds__(128)
void cos_final_kernel(const float* __restrict__ partial, int nparts,
                      float* __restrict__ out, float scale)
{
    float s = 0.f;
    for (int i = threadIdx.x; i < nparts; i += 128) s += partial[i];
    __shared__ float red[128];
    red[threadIdx.x] = s; __syncthreads();
    for (int off = 64; off > 0; off >>= 1) {
        if (threadIdx.x < off) red[threadIdx.x] += red[threadIdx.x + off];
        __syncthreads();
    }
    if (threadIdx.x == 0) *out = red[0] * scale;
}

// ---------------------------------------------------------------------------
// Host-side dispatch: tiled kernel for 64-aligned shapes, simple otherwise.
// ---------------------------------------------------------------------------
static void run_gemm(bool ta, bool tb, const float* A, const float* Bp,
                     const float* Cin, float* D,
                     int M, int N, int K, int lda, int ldb, int ldci, int ldco,
                     long long sA, long long sB, long long sCi, long long sCo,
                     float alpha, float beta, float diag, int batch,
                     hipStream_t st)
{
    const bool tiled = (M % BM == 0) && (N % BN == 0) && (K % BK == 0);
    if (tiled) {
        dim3 grid(N / BN, M / BM, batch), block(128);
        if (!ta && !tb)
            gemm_wmma_tiled<false, false><<<grid, block, 0, st>>>(A, Bp, Cin, D,
                M, N, K, lda, ldb, ldci, ldco, sA, sB, sCi, sCo, alpha, beta, diag);
        else if (ta && !tb)
            gemm_wmma_tiled<true, false><<<grid, block, 0, st>>>(A, Bp, Cin, D,
                M, N, K, lda, ldb, ldci, ldco, sA, sB, sCi, sCo, alpha, beta, diag);
        else if (!ta && tb)
            gemm_wmma_tiled<false, true><<<grid, block, 0, st>>>(A, Bp, Cin, D,
                M, N, K, lda, ldb, ldci, ldco, sA, sB, sCi, sCo, alpha, beta, diag);
        else
            gemm_wmma_tiled<true, true><<<grid, block, 0, st>>>(A, Bp, Cin, D,
                M, N, K, lda, ldb, ldci, ldco, sA, sB, sCi, sCo, alpha, beta, diag);
    } else {
        dim3 grid(N / 16, M / 16, batch), block(32);
        if (!ta && !tb)
            gemm_wmma_simple<false, false><<<grid, block, 0, st>>>(A, Bp, Cin, D,
                M, N, K, lda, ldb, ldci, ldco, sA, sB, sCi, sCo, alpha, beta, diag);
        else if (ta && !tb)
            gemm_wmma_simple<true, false><<<grid, block, 0, st>>>(A, Bp, Cin, D,
                M, N, K, lda, ldb, ldci, ldco, sA, sB, sCi, sCo, alpha, beta, diag);
        else if (!ta && tb)
            gemm_wmma_simple<false, true><<<grid, block, 0, st>>>(A, Bp, Cin, D,
                M, N, K, lda, ldb, ldci, ldco, sA, sB, sCi, sCo, alpha, beta, diag);
        else
            gemm_wmma_simple<true, true><<<grid, block, 0, st>>>(A, Bp, Cin, D,
                M, N, K, lda, ldb, ldci, ldco, sA, sB, sCi, sCo, alpha, beta, diag);
    }
}

extern "C" void kernel_launch(void* const* d_in, const int* in_sizes, int n_in,
                              void* d_out, int out_size, void* d_ws, size_t ws_size,
                              hipStream_t stream)
{
    (void)in_sizes; (void)n_in; (void)out_size; (void)ws_size;
    const int B = 64, n = 512, N = 384, RANK = 24, RP = 32; // rank padded 24->32
    const int NS_ITERS = 5, SUB_ITERS = 8, POLAR_ITERS = 16;
    const float EPS = 1e-5f;

    const float* src = (const float*)d_in[0];
    const float* tgt = (const float*)d_in[1];
    float* out = (float*)d_out;

    const long long sXN = (long long)n * N;    // 512x384
    const long long sNN = (long long)N * N;    // 384x384
    const long long sNR = (long long)N * RP;   // 384x32
    const long long snR = (long long)n * RP;   // 512x32
    const long long sRR = (long long)RP * RP;  // 32x32

    float* ws = (float*)d_ws;
    long long o = 0;
    auto alloc = [&](long long cnt) { float* p = ws + o; o += cnt; return p; };
    float* Sw = alloc((long long)B * sXN);
    float* Tw = alloc((long long)B * sXN);
    float* M1 = alloc((long long)B * sNN);
    float* M2 = alloc((long long)B * sNN);
    float* M3 = alloc((long long)B * sNN);
    float* M4 = alloc((long long)B * sNN);
    float* M5 = alloc((long long)B * sNN);
    float* Q  = alloc((long long)B * sNR);
    float* Zq = alloc((long long)B * sNR);
    float* Sp = alloc((long long)B * snR);
    float* Tp = alloc((long long)B * snR);
    float* Ub = alloc((long long)B * snR);
    float* s1 = alloc((long long)B * sRR);
    float* s2 = alloc((long long)B * sRR);
    float* s3 = alloc((long long)B * sRR);
    float* s4 = alloc((long long)B * sRR);
    float* s5 = alloc((long long)B * sRR);
    float* nrm  = alloc(B);
    float* part = alloc(256);

    // Newton-Schulz inverse square root on batched dim x dim SPD matrices.
    auto ns_invsqrt = [&](float* Cb, float* Yb, float* Zb, float* Tb, float* Y2b,
                          int dim, int iters) -> float* {
        const long long st  = (long long)dim * dim;
        const long long tot = (long long)B * st;
        const int gb = (int)((tot + 255) / 256);
        frob_kernel<<<B, 256, 0, stream>>>(Cb, st, (int)st, nrm);
        scale_kernel<<<gb, 256, 0, stream>>>(Cb, Yb, nrm, 0, tot, (int)st);
        set_diag_kernel<<<gb, 256, 0, stream>>>(Zb, 1.f, dim, tot, (int)st);
        float *pY = Yb, *pZ = Zb, *pY2 = Y2b, *pZ2 = Cb;
        for (int it = 0; it < iters; ++it) {
            run_gemm(false, false, pZ, pY, Tb, Tb, dim, dim, dim,
                     dim, dim, dim, dim, st, st, st, st, -0.5f, 0.f, 1.5f, B, stream);
            run_gemm(false, false, pY, Tb, Tb, pY2, dim, dim, dim,
                     dim, dim, dim, dim, st, st, st, st, 1.f, 0.f, 0.f, B, stream);
            run_gemm(false, false, Tb, pZ, Tb, pZ2, dim, dim, dim,
                     dim, dim, dim, dim, st, st, st, st, 1.f, 0.f, 0.f, B, stream);
            float* t0 = pY; pY = pY2; pY2 = t0;
            float* t1 = pZ; pZ = pZ2; pZ2 = t1;
        }
        scale_kernel<<<gb, 256, 0, stream>>>(pZ, Tb, nrm, 1, tot, (int)st);
        return Tb;
    };

    auto ortho = [&](float* Zin, float* Qout, int dim) {
        run_gemm(true, false, Zin, Zin, s1, s1, RP, RP, dim,
                 RP, RP, RP, RP, (long long)dim * RP, (long long)dim * RP, sRR, sRR,
                 1.f, 0.f, 0.f, B, stream);
        float* W = ns_invsqrt(s1, s2, s3, s4, s5, RP, 5);
        run_gemm(false, false, Zin, W, Qout, Qout, dim, RP, RP,
                 RP, RP, RP, RP, (long long)dim * RP, sRR, (long long)dim * RP,
                 (long long)dim * RP, 1.f, 0.f, 0.f, B, stream);
    };

    auto whiten = [&](const float* X, float* Xw) {
        run_gemm(true, false, X, X, M1, M1, N, N, n,
                 N, N, N, N, sXN, sXN, sNN, sNN, 1.0f / (float)n, 0.f, EPS, B, stream);
        float* W = ns_invsqrt(M1, M2, M3, M4, M5, N, NS_ITERS);
        run_gemm(false, false, X, W, Xw, Xw, n, N, N,
                 N, N, N, N, sXN, sNN, sXN, sXN, 1.f, 0.f, 0.f, B, stream);
    };

    // 1) Whiten both inputs.
    whiten(src, Sw);
    whiten(tgt, Tw);

    // 2) G = S^T S + T^T T
    run_gemm(true, false, Sw, Sw, M1, M1, N, N, n,
             N, N, N, N, sXN, sXN, sNN, sNN, 1.f, 0.f, 0.f, B, stream);
    run_gemm(true, false, Tw, Tw, M1, M1, N, N, n,
             N, N, N, N, sXN, sXN, sNN, sNN, 1.f, 1.f, 0.f, B, stream);

    // 3) Subspace iteration for Q (B, N, RP) — pad columns stay exactly zero.
    qinit_kernel<<<(int)(((long long)B * sNR + 255) / 256), 256, 0, stream>>>(
        Zq, (long long)B * sNR, RP, RANK);
    ortho(Zq, Q, N);
    for (int it = 0; it < SUB_ITERS; ++it) {
        run_gemm(false, false, M1, Q, Zq, Zq, N, RP, N,
                 N, RP, RP, RP, sNN, sNR, sNR, sNR, 1.f, 0.f, 0.f, B, stream);
        ortho(Zq, Q, N);
    }

    // 4) Project: Sp = S@Q, Tp = T@Q; M = Sp^T Tp
    run_gemm(false, false, Sw, Q, Sp, Sp, n, RP, N,
             N, RP, RP, RP, sXN, sNR, snR, snR, 1.f, 0.f, 0.f, B, stream);
    run_gemm(false, false, Tw, Q, Tp, Tp, n, RP, N,
             N, RP, RP, RP, sXN, sNR, snR, snR, 1.f, 0.f, 0.f, B, stream);
    run_gemm(true, false, Sp, Tp, s1, s1, RP, RP, n,
             RP, RP, RP, RP, snR, snR, sRR, sRR, 1.f, 0.f, 0.f, B, stream);

    // 5) Polar factor via Newton-Schulz: X <- 0.5 X (3I - X^T X)
    {
        const long long tot = (long long)B * sRR;
        const int gb = (int)((tot + 255) / 256);
        frob_kernel<<<B, 256, 0, stream>>>(s1, sRR, (int)sRR, nrm);
        scale_kernel<<<gb, 256, 0, stream>>>(s1, s2, nrm, 0, tot, (int)sRR);
        float *pX = s2, *pX2 = s4;
        for (int it = 0; it < POLAR_ITERS; ++it) {
            run_gemm(true, false, pX, pX, s3, s3, RP, RP, RP,
                     RP, RP, RP, RP, sRR, sRR, sRR, sRR, -0.5f, 0.f, 1.5f, B, stream);
            run_gemm(false, false, pX, s3, s3, pX2, RP, RP, RP,
                     RP, RP, RP, RP, sRR, sRR, sRR, sRR, 1.f, 0.f, 0.f, B, stream);
            float* t = pX; pX = pX2; pX2 = t;
        }
        shift_diag_kernel<<<gb, 256, 0, stream>>>(pX, s5, -1.f, RP, tot, (int)sRR);
    }

    // 6) aligned = S + Sp @ E @ Q^T
    run_gemm(false, false, Sp, s5, Ub, Ub, n, RP, RP,
             RP, RP, RP, RP, snR, sRR, snR, snR, 1.f, 0.f, 0.f, B, stream);
    hipMemcpyAsync(out, Sw, sizeof(float) * (size_t)B * (size_t)sXN,
                   hipMemcpyDeviceToDevice, stream);
    run_gemm(false, true, Ub, Q, out, out, n, N, RP,
             RP, RP, N, N, snR, sNR, sXN, sXN, 1.f, 1.f, 0.f, B, stream);

    // 7) cos_after diagnostic (deterministic two-stage reduction)
    const long long rows = (long long)B * n;
    cos_partial_kernel<<<128, 256, 0, stream>>>(out, tgt, part, rows, N);
    cos_final_kernel<<<1, 128, 0, stream>>>(part, 128, out + (long long)B * sXN,
                                            1.0f / (float)rows);
}